// MixHop_59682865545364
// MI455X (gfx1250) — compile-verified
//
#include <hip/hip_runtime.h>
#include <hip/hip_bf16.h>

typedef float v2f __attribute__((ext_vector_type(2)));
typedef float v8f __attribute__((ext_vector_type(8)));

#define NFEAT 512
#define NHID 128
#define NCLASS 64

// ---------------------------------------------------------------------------
// CSR construction
// ---------------------------------------------------------------------------
__global__ void zero_i32_k(int* __restrict__ p, int n) {
    int i = blockIdx.x * blockDim.x + threadIdx.x;
    if (i < n) p[i] = 0;
}

__global__ void hist_k(const int* __restrict__ row, int E, int* __restrict__ deg) {
    int e = blockIdx.x * blockDim.x + threadIdx.x;
    if (e < E) atomicAdd(&deg[row[e]], 1);
}

__global__ __launch_bounds__(1024) void scan_k(const int* __restrict__ deg, int N,
                                               int* __restrict__ rowptr,
                                               int* __restrict__ cursor) {
    __shared__ int s[1024];
    int t = threadIdx.x;
    int running = 0;
    for (int base = 0; base < N; base += 1024) {
        int i = base + t;
        int v = (i < N) ? deg[i] : 0;
        s[t] = v;
        __syncthreads();
        for (int off = 1; off < 1024; off <<= 1) {
            int x = (t >= off) ? s[t - off] : 0;
            __syncthreads();
            s[t] += x;
            __syncthreads();
        }
        int excl = running + s[t] - v;
        if (i < N) { rowptr[i] = excl; cursor[i] = excl; }
        running += s[1023];
        __syncthreads();
    }
    if (t == 0) rowptr[N] = running;
}

__global__ void scatter_k(const int* __restrict__ row, const int* __restrict__ col,
                          const float* __restrict__ vals, int E,
                          int* __restrict__ cursor,
                          int* __restrict__ es_col, float* __restrict__ es_val) {
    int e = blockIdx.x * blockDim.x + threadIdx.x;
    if (e < E) {
        int r = row[e];
        int p = atomicAdd(&cursor[r], 1);
        es_col[p] = col[e];
        es_val[p] = vals[e];
    }
}

// ---------------------------------------------------------------------------
// FP32 WMMA GEMM:  C[N_rows, ncols] = A[N_rows,K] @ W.T + b
// mode 0: ncols=384, W split {W0,W1,W2}; cols<128 -> relu -> X[.,col] (ld 384)
//         cols>=128 -> Z12[., col-128] (ld 256), no relu
// mode 1: ncols=64, W=W_out; -> out0[., col] (ld 64), no relu
// Block: 256 threads (8 waves), tile 128 rows x 32 cols, K-chunks of 32 in LDS.
// As: row-major [row][k] (stride 34: 8B-aligned even-k fragment reads)
// Bs: col-major [col][k] (stride 36: 16B-aligned float4 stores, contiguous
//     k-pairs -> B fragments are single ds_load_b64, no v_mov packing)
// ---------------------------------------------------------------------------
__global__ __launch_bounds__(256) void gemm_wmma_k(
    const float* __restrict__ A, int lda, int K, int N_rows,
    const float* __restrict__ W0, const float* __restrict__ W1, const float* __restrict__ W2,
    const float* __restrict__ b0, const float* __restrict__ b1, const float* __restrict__ b2,
    float* __restrict__ out0, float* __restrict__ out1, int mode)
{
    __shared__ float As[128][34];
    __shared__ float Bs[32][36];

    const int tid  = threadIdx.x;
    const int lane = tid & 31;
    const int w    = tid >> 5;
    const int r0   = blockIdx.x * 128;
    const int c0   = blockIdx.y * 32;

    v8f acc0 = {};
    v8f acc1 = {};

    const int m  = lane & 15;
    const int kb = (lane >> 4) * 2;

    for (int kc = 0; kc < K; kc += 32) {
        // ---- load A tile (128 rows x 32 k) ----
        {
            int rr = tid >> 3;            // 0..31
            int cc = (tid & 7) * 4;       // 0..28
            #pragma unroll
            for (int i = 0; i < 4; ++i) {
                int rowl = rr + i * 32;
                int grow = r0 + rowl;
                if (grow >= N_rows) grow = N_rows - 1;   // clamp; store is guarded
                const float4 v = *(const float4*)(A + (size_t)grow * lda + kc + cc);
                As[rowl][cc]     = v.x;
                As[rowl][cc + 1] = v.y;
                As[rowl][cc + 2] = v.z;
                As[rowl][cc + 3] = v.w;
            }
        }
        // ---- load B tile: Wt[kc..kc+31, c0..c0+31]; W is [ncols_part, K] row-major
        //      so a column of Wt is contiguous in k -> float4 load, float4 store.
        {
            int j  = tid >> 3;            // 0..31 (column within tile)
            int k4 = (tid & 7) * 4;
            int gcol = c0 + j;
            int sel  = gcol >> 7;
            const float* Wp = (sel == 0) ? W0 : ((sel == 1) ? W1 : W2);
            int jj = gcol & 127;
            const float4 v = *(const float4*)(Wp + (size_t)jj * K + kc + k4);
            *(float4*)&Bs[j][k4] = v;
        }
        __syncthreads();

        const int Ms = w * 16;
        #pragma unroll
        for (int kk = 0; kk < 32; kk += 4) {
            const v2f a  = *(const v2f*)&As[Ms + m][kk + kb];
            const v2f bA = *(const v2f*)&Bs[m][kk + kb];
            acc0 = __builtin_amdgcn_wmma_f32_16x16x4_f32(false, a, false, bA,
                                                         (short)0, acc0, false, false);
            const v2f bB = *(const v2f*)&Bs[16 + m][kk + kb];
            acc1 = __builtin_amdgcn_wmma_f32_16x16x4_f32(false, a, false, bB,
                                                         (short)0, acc1, false, false);
        }
        __syncthreads();
    }

    // ---- epilogue: lane l holds C[r + 8*(l>>4)][l&15] in acc[r] ----
    const int half = lane >> 4;
    #pragma unroll
    for (int sub = 0; sub < 2; ++sub) {
        v8f acc = (sub == 0) ? acc0 : acc1;
        int gcol = c0 + sub * 16 + m;
        int sel  = gcol >> 7;
        const float* bp = (sel == 0) ? b0 : ((sel == 1) ? b1 : b2);
        float bias = bp[gcol & 127];
        #pragma unroll
        for (int r = 0; r < 8; ++r) {
            int grow = r0 + w * 16 + r + half * 8;
            if (grow < N_rows) {
                float v = acc[r] + bias;
                if (mode == 0) {
                    if (gcol < 128)
                        out0[(size_t)grow * 384 + gcol] = fmaxf(v, 0.0f);      // relu(h0)
                    else
                        out1[(size_t)grow * 256 + (gcol - 128)] = v;           // Z1|Z2
                } else {
                    out0[(size_t)grow * 64 + gcol] = v;                        // Y
                }
            }
        }
    }
}

// ---------------------------------------------------------------------------
// Mid SPMM (row-per-wave, CSR): X[:,128:256]=relu(A·Z1), X[:,256:384]=relu(A²·Z2)
// Edge-list loop runs on scalar loads (readfirstlane'd bounds).
// ---------------------------------------------------------------------------
__global__ __launch_bounds__(256) void spmm_mid_k(
    const int* __restrict__ rowptr, const int* __restrict__ es_col,
    const float* __restrict__ es_val, const float* __restrict__ Z12,
    float* __restrict__ X, int N)
{
    int r = blockIdx.x * 8 + (threadIdx.x >> 5);
    if (r >= N) return;                      // wave-uniform exit
    int lane = threadIdx.x & 31;
    int start = __builtin_amdgcn_readfirstlane(rowptr[r]);
    int end   = __builtin_amdgcn_readfirstlane(rowptr[r + 1]);

    float4 a1 = {0.f, 0.f, 0.f, 0.f};
    float4 a2 = {0.f, 0.f, 0.f, 0.f};
    for (int j = start; j < end; ++j) {      // j in SGPR -> scalar edge loads
        int   c = es_col[j];
        float v = es_val[j];
        float v2 = v * v;
        const float4 z1 = *(const float4*)(Z12 + (size_t)c * 256 + lane * 4);
        const float4 z2 = *(const float4*)(Z12 + (size_t)c * 256 + 128 + lane * 4);
        a1.x += v * z1.x;  a1.y += v * z1.y;  a1.z += v * z1.z;  a1.w += v * z1.w;
        a2.x += v2 * z2.x; a2.y += v2 * z2.y; a2.z += v2 * z2.z; a2.w += v2 * z2.w;
    }
    float4 o1 = {fmaxf(a1.x, 0.f), fmaxf(a1.y, 0.f), fmaxf(a1.z, 0.f), fmaxf(a1.w, 0.f)};
    float4 o2 = {fmaxf(a2.x, 0.f), fmaxf(a2.y, 0.f), fmaxf(a2.z, 0.f), fmaxf(a2.w, 0.f)};
    float* xr = X + (size_t)r * 384;
    *(float4*)(xr + 128 + lane * 4) = o1;
    *(float4*)(xr + 256 + lane * 4) = o2;
}

// ---------------------------------------------------------------------------
// Final SPMM + fused log_softmax: out[r] = log_softmax(sum vals*Y[col])
// ---------------------------------------------------------------------------
__global__ __launch_bounds__(256) void spmm_out_k(
    const int* __restrict__ rowptr, const int* __restrict__ es_col,
    const float* __restrict__ es_val, const float* __restrict__ Y,
    float* __restrict__ out, int N)
{
    int r = blockIdx.x * 8 + (threadIdx.x >> 5);
    if (r >= N) return;
    int lane = threadIdx.x & 31;
    int start = __builtin_amdgcn_readfirstlane(rowptr[r]);
    int end   = __builtin_amdgcn_readfirstlane(rowptr[r + 1]);

    float ax = 0.f, ay = 0.f;
    for (int j = start; j < end; ++j) {
        int   c = es_col[j];
        float v = es_val[j];
        const float2 y = *(const float2*)(Y + (size_t)c * 64 + lane * 2);
        ax += v * y.x;
        ay += v * y.y;
    }
    // log_softmax over the 64 row values (2 per lane, 32 lanes)
    float mx = fmaxf(ax, ay);
    #pragma unroll
    for (int off = 16; off >= 1; off >>= 1)
        mx = fmaxf(mx, __shfl_xor(mx, off, 32));
    float s = expf(ax - mx) + expf(ay - mx);
    #pragma unroll
    for (int off = 16; off >= 1; off >>= 1)
        s += __shfl_xor(s, off, 32);
    float lse = mx + logf(s);
    float2 o = {ax - lse, ay - lse};
    *(float2*)(out + (size_t)r * 64 + lane * 2) = o;
}

// ---------------------------------------------------------------------------
extern "C" void kernel_launch(void* const* d_in, const int* in_sizes, int n_in,
                              void* d_out, int out_size, void* d_ws, size_t ws_size,
                              hipStream_t stream) {
    const float* features = (const float*)d_in[0];
    const int*   edge_row = (const int*)d_in[1];
    const int*   edge_col = (const int*)d_in[2];
    const float* edge_val = (const float*)d_in[3];
    const float* W0 = (const float*)d_in[4];
    const float* b0 = (const float*)d_in[5];
    const float* W1 = (const float*)d_in[6];
    const float* b1 = (const float*)d_in[7];
    const float* W2 = (const float*)d_in[8];
    const float* b2 = (const float*)d_in[9];
    const float* W_out = (const float*)d_in[10];
    const float* b_out = (const float*)d_in[11];
    float* out = (float*)d_out;

    const int N = in_sizes[0] / NFEAT;
    const int E = in_sizes[1];

    // ---- workspace carve-out (256B aligned) ----
    char* ws = (char*)d_ws;
    size_t off = 0;
    auto carve = [&](size_t bytes) -> char* {
        char* p = ws + off;
        off = (off + bytes + 255) & ~(size_t)255;
        return p;
    };
    float* X      = (float*)carve((size_t)N * 384 * 4);   // [N,384] relu'd concat
    float* Z12    = (float*)carve((size_t)N * 256 * 4);   // [N,256] pre-relu Z1|Z2
    float* Y      = Z12;                                  // [N,64]; alias (Z12 dead by then)
    int*   rowptr = (int*)carve((size_t)(N + 1) * 4);
    int*   cursor = (int*)carve((size_t)N * 4);
    int*   deg    = (int*)carve((size_t)N * 4);
    int*   es_col = (int*)carve((size_t)E * 4);
    float* es_val = (float*)carve((size_t)E * 4);
    (void)ws_size; (void)n_in; (void)out_size;

    // ---- CSR build ----
    zero_i32_k<<<(N + 255) / 256, 256, 0, stream>>>(deg, N);
    hist_k<<<(E + 255) / 256, 256, 0, stream>>>(edge_row, E, deg);
    scan_k<<<1, 1024, 0, stream>>>(deg, N, rowptr, cursor);
    scatter_k<<<(E + 255) / 256, 256, 0, stream>>>(edge_row, edge_col, edge_val, E,
                                                   cursor, es_col, es_val);

    // ---- GEMM 1: [h0|Z1|Z2] = features @ [W0;W1;W2].T + b ----
    {
        dim3 grid((N + 127) / 128, 12);
        gemm_wmma_k<<<grid, 256, 0, stream>>>(features, NFEAT, NFEAT, N,
                                              W0, W1, W2, b0, b1, b2,
                                              X, Z12, 0);
    }
    // ---- SPMM mid: X[:,128:384] = relu(A·Z1 | A²·Z2) ----
    spmm_mid_k<<<(N + 7) / 8, 256, 0, stream>>>(rowptr, es_col, es_val, Z12, X, N);

    // ---- GEMM 2: Y = X @ W_out.T + b_out ----
    {
        dim3 grid((N + 127) / 128, 2);
        gemm_wmma_k<<<grid, 256, 0, stream>>>(X, 384, 384, N,
                                              W_out, W_out, W_out, b_out, b_out, b_out,
                                              Y, nullptr, 1);
    }
    // ---- SPMM out + log_softmax ----
    spmm_out_k<<<(N + 7) / 8, 256, 0, stream>>>(rowptr, es_col, es_val, Y, out, N);
}